// PoseMetrics_51402168599325
// MI455X (gfx1250) — compile-verified
//
#include <hip/hip_runtime.h>
#include <stdint.h>

// ---------------- configuration ----------------
#define JOINTS 14
#define TILE 32                         // samples per wave-tile (== wave32 lanes)
#define WPB 2                           // waves per block
#define BLOCK (WPB * 32)                // 64 threads
#define NBLOCKS 2048
#define SAMPLE_FLOATS (3 * JOINTS)      // 42
#define SAMPLE_BYTES (SAMPLE_FLOATS * 4)// 168
#define TILE_FLOATS (TILE * SAMPLE_FLOATS) // 1344
#define TILE_BYTES (TILE * SAMPLE_BYTES)   // 5376
#define TDM_ELEMS (TILE_BYTES / 8)         // 672 x 8-byte elements per array tile

typedef uint32_t v4u __attribute__((ext_vector_type(4)));
typedef uint32_t v8u __attribute__((ext_vector_type(8)));

// ---------------- CDNA5 Tensor Data Mover (TDM) ----------------
// One wave-level DMA descriptor moves a whole 5376B contiguous tile into LDS.
// D# group1 (constant): data_size=8B, 1-D tile: tensor_dim0=tile_dim0=672,
// tile_dim1=tile_dim2=1, no pad/iterate/multicast (not in a cluster).
__device__ __forceinline__ v8u tdm_group1() {
    v8u g1;
    g1[0] = 0x00030000u;                  // workgroup_mask=0, data_size=3 (8B)
    g1[1] = (uint32_t)TDM_ELEMS << 16;    // [31:16] = tensor_dim0[15:0]
    g1[2] = 1u << 16;                     // [15:0]=tensor_dim0[31:16]=0, [31:16]=tensor_dim1 lo = 1
    g1[3] = (uint32_t)TDM_ELEMS << 16;    // [15:0]=tensor_dim1 hi=0, [31:16]=tile_dim0=672
    g1[4] = 0x00010001u;                  // tile_dim1=1, tile_dim2=1
    g1[5] = (uint32_t)TDM_ELEMS;          // tensor_dim0_stride lo32 = 672
    g1[6] = (uint32_t)TDM_ELEMS << 16;    // stride0 hi16=0, tensor_dim1_stride lo16 = 672
    g1[7] = 0u;                           // tensor_dim1_stride hi = 0
    return g1;
}

__device__ __forceinline__ void tdm_load(uint64_t gaddr, uint32_t lds_addr, v8u g1) {
    v4u g0;
    g0[0] = 1u;                                   // count=1, user descriptor, no gather
    g0[1] = lds_addr;                             // LDS byte address
    g0[2] = (uint32_t)gaddr;                      // global_addr[31:0]
    g0[3] = (uint32_t)((gaddr >> 32) & 0x01FFFFFFu) | 0x80000000u; // addr[56:32] | type=2
    asm volatile("tensor_load_to_lds %0, %1" :: "s"(g0), "s"(g1) : "memory");
}

__device__ __forceinline__ void tdm_issue_tile(uint64_t predAddr, uint64_t targAddr,
                                               uint32_t ldsP, uint32_t ldsT, v8u g1) {
    tdm_load(predAddr, ldsP, g1);
    tdm_load(targAddr, ldsT, g1);
}

// eigenvector of symmetric 3x3 (a..) for eigenvalue lam, via row cross-products
__device__ __forceinline__ void eig_vec(float a00, float a01, float a02,
                                        float a11, float a12, float a22,
                                        float lam, float& vx, float& vy, float& vz) {
    float m00 = a00 - lam, m11 = a11 - lam, m22 = a22 - lam;
    // r0=(m00,a01,a02) r1=(a01,m11,a12) r2=(a02,a12,m22)
    float x0 = a01 * a12 - a02 * m11, y0 = a02 * a01 - m00 * a12, z0 = m00 * m11 - a01 * a01;
    float x1 = a01 * m22 - a02 * a12, y1 = a02 * a02 - m00 * m22, z1 = m00 * a12 - a01 * a02;
    float x2 = m11 * m22 - a12 * a12, y2 = a12 * a02 - a01 * m22, z2 = a01 * a12 - m11 * a02;
    float n0 = x0 * x0 + y0 * y0 + z0 * z0;
    float n1 = x1 * x1 + y1 * y1 + z1 * z1;
    float n2 = x2 * x2 + y2 * y2 + z2 * z2;
    float x = x0, y = y0, z = z0, n = n0;
    if (n1 > n) { x = x1; y = y1; z = z1; n = n1; }
    if (n2 > n) { x = x2; y = y2; z = z2; n = n2; }
    float inv = rsqrtf(fmaxf(n, 1e-30f));
    vx = x * inv; vy = y * inv; vz = z * inv;
}

__device__ __forceinline__ void compute_tile(const float* __restrict__ P,
                                             const float* __restrict__ T,
                                             int lane,
                                             float& accM, float& accP, float& accA) {
    const float* p = P + lane * SAMPLE_FLOATS;   // [3][14] row-major
    const float* t = T + lane * SAMPLE_FLOATS;

    float h00 = 0, h01 = 0, h02 = 0, h10 = 0, h11 = 0, h12 = 0, h20 = 0, h21 = 0, h22 = 0;
    float pm0 = 0, pm1 = 0, pm2 = 0, tm0 = 0, tm1 = 0, tm2 = 0;
    float sm = 0, sa = 0;
    float q0 = 0, q1 = 0, q2 = 0;   // p at j-1
    float w0 = 0, w1 = 0, w2 = 0;   // p at j-2

#pragma unroll
    for (int j = 0; j < JOINTS; ++j) {
        float p0 = p[j], p1 = p[JOINTS + j], p2 = p[2 * JOINTS + j];
        float t0 = t[j], t1 = t[JOINTS + j], t2 = t[2 * JOINTS + j];
        float d0 = p0 - t0, d1 = p1 - t1, d2 = p2 - t2;
        sm += sqrtf(d0 * d0 + d1 * d1 + d2 * d2);
        pm0 += p0; pm1 += p1; pm2 += p2;
        tm0 += t0; tm1 += t1; tm2 += t2;
        h00 += p0 * t0; h01 += p0 * t1; h02 += p0 * t2;
        h10 += p1 * t0; h11 += p1 * t1; h12 += p1 * t2;
        h20 += p2 * t0; h21 += p2 * t1; h22 += p2 * t2;
        if (j >= 2) {
            float a0 = p0 - 2.0f * q0 + w0;
            float a1 = p1 - 2.0f * q1 + w1;
            float a2 = p2 - 2.0f * q2 + w2;
            sa += sqrtf(a0 * a0 + a1 * a1 + a2 * a2);
        }
        w0 = q0; w1 = q1; w2 = q2;
        q0 = p0; q1 = p1; q2 = p2;
    }

    const float invJ = 1.0f / (float)JOINTS;
    pm0 *= invJ; pm1 *= invJ; pm2 *= invJ;
    tm0 *= invJ; tm1 *= invJ; tm2 *= invJ;
    const float J = (float)JOINTS;
    h00 -= J * pm0 * tm0; h01 -= J * pm0 * tm1; h02 -= J * pm0 * tm2;
    h10 -= J * pm1 * tm0; h11 -= J * pm1 * tm1; h12 -= J * pm1 * tm2;
    h20 -= J * pm2 * tm0; h21 -= J * pm2 * tm1; h22 -= J * pm2 * tm2;

    float detH = h00 * (h11 * h22 - h12 * h21)
               - h01 * (h10 * h22 - h12 * h20)
               + h02 * (h10 * h21 - h11 * h20);

    // A = H^T H (symmetric PSD)
    float a00 = h00 * h00 + h10 * h10 + h20 * h20;
    float a01 = h00 * h01 + h10 * h11 + h20 * h21;
    float a02 = h00 * h02 + h10 * h12 + h20 * h22;
    float a11 = h01 * h01 + h11 * h11 + h21 * h21;
    float a12 = h01 * h02 + h11 * h12 + h21 * h22;
    float a22 = h02 * h02 + h12 * h12 + h22 * h22;

    // closed-form eigenvalues (descending l1 >= l2 >= l3)
    float qv = (a00 + a11 + a22) * (1.0f / 3.0f);
    float b00 = a00 - qv, b11 = a11 - qv, b22 = a22 - qv;
    float p2s = b00 * b00 + b11 * b11 + b22 * b22
              + 2.0f * (a01 * a01 + a02 * a02 + a12 * a12);
    float pp = sqrtf(p2s * (1.0f / 6.0f)) + 1e-20f;
    float invp = 1.0f / pp;
    float detB = b00 * (b11 * b22 - a12 * a12)
               - a01 * (a01 * b22 - a12 * a02)
               + a02 * (a01 * a12 - b11 * a02);
    float rr = 0.5f * detB * invp * invp * invp;
    rr = fminf(1.0f, fmaxf(-1.0f, rr));
    float phi = acosf(rr) * (1.0f / 3.0f);
    float l1 = qv + 2.0f * pp * __cosf(phi);
    float l3 = qv + 2.0f * pp * __cosf(phi + 2.0943951023931953f);
    float l2 = 3.0f * qv - l1 - l3;

    float v1x, v1y, v1z, v3x, v3y, v3z;
    eig_vec(a00, a01, a02, a11, a12, a22, l1, v1x, v1y, v1z);
    eig_vec(a00, a01, a02, a11, a12, a22, l3, v3x, v3y, v3z);
    float v2x = v3y * v1z - v3z * v1y;
    float v2y = v3z * v1x - v3x * v1z;
    float v2z = v3x * v1y - v3y * v1x;
    float n2i = rsqrtf(fmaxf(v2x * v2x + v2y * v2y + v2z * v2z, 1e-30f));
    v2x *= n2i; v2y *= n2i; v2z *= n2i;

    const float eps = 1e-12f;
    float c1 = rsqrtf(fmaxf(l1, eps));
    float c2 = rsqrtf(fmaxf(l2, eps));
    float c3 = rsqrtf(fmaxf(l3, eps));
    if (detH < 0.0f) c3 = -c3;          // Kabsch reflection fix

    // W = V diag(c1,c2,c3) V^T  (symmetric);  R = W H^T
    float w00s = c1 * v1x * v1x + c2 * v2x * v2x + c3 * v3x * v3x;
    float w01s = c1 * v1x * v1y + c2 * v2x * v2y + c3 * v3x * v3y;
    float w02s = c1 * v1x * v1z + c2 * v2x * v2z + c3 * v3x * v3z;
    float w11s = c1 * v1y * v1y + c2 * v2y * v2y + c3 * v3y * v3y;
    float w12s = c1 * v1y * v1z + c2 * v2y * v2z + c3 * v3y * v3z;
    float w22s = c1 * v1z * v1z + c2 * v2z * v2z + c3 * v3z * v3z;

    float r00 = w00s * h00 + w01s * h01 + w02s * h02;
    float r01 = w00s * h10 + w01s * h11 + w02s * h12;
    float r02 = w00s * h20 + w01s * h21 + w02s * h22;
    float r10 = w01s * h00 + w11s * h01 + w12s * h02;
    float r11 = w01s * h10 + w11s * h11 + w12s * h12;
    float r12 = w01s * h20 + w11s * h21 + w12s * h22;
    float r20 = w02s * h00 + w12s * h01 + w22s * h02;
    float r21 = w02s * h10 + w12s * h11 + w22s * h12;
    float r22 = w02s * h20 + w12s * h21 + w22s * h22;

    // PA-MPJPE pass:  || R*pred_c - targ_c ||
    float sp = 0;
#pragma unroll
    for (int j = 0; j < JOINTS; ++j) {
        float pc0 = p[j] - pm0, pc1 = p[JOINTS + j] - pm1, pc2 = p[2 * JOINTS + j] - pm2;
        float tc0 = t[j] - tm0, tc1 = t[JOINTS + j] - tm1, tc2 = t[2 * JOINTS + j] - tm2;
        float e0 = r00 * pc0 + r01 * pc1 + r02 * pc2 - tc0;
        float e1 = r10 * pc0 + r11 * pc1 + r12 * pc2 - tc1;
        float e2 = r20 * pc0 + r21 * pc1 + r22 * pc2 - tc2;
        sp += sqrtf(e0 * e0 + e1 * e1 + e2 * e2);
    }

    accM += sm; accP += sp; accA += sa;
}

__global__ __launch_bounds__(BLOCK) void pose_main(const float* __restrict__ pred,
                                                   const float* __restrict__ targ,
                                                   float* __restrict__ ws, int nTiles) {
    __shared__ float tiles[WPB][2][2 * TILE_FLOATS]; // [wave][buf][pred|targ]
    __shared__ float red[WPB][3];

    const int tid = threadIdx.x;
    const int lane = tid & 31;
    // wave id is wave-uniform; force it scalar so all TDM descriptor math is SALU
    const int wave = __builtin_amdgcn_readfirstlane(tid >> 5);
    const int gwave = blockIdx.x * WPB + wave;
    const int stride = gridDim.x * WPB;

    uint32_t ldsP[2], ldsT[2];
#pragma unroll
    for (int b = 0; b < 2; ++b) {
        ldsP[b] = (uint32_t)(uintptr_t)&tiles[wave][b][0];          // low 32 bits == LDS offset
        ldsT[b] = (uint32_t)(uintptr_t)&tiles[wave][b][TILE_FLOATS];
    }

    const v8u g1 = tdm_group1();
    const uint64_t baseP = (uint64_t)(uintptr_t)pred;
    const uint64_t baseT = (uint64_t)(uintptr_t)targ;

    float accM = 0.0f, accP = 0.0f, accA = 0.0f;

    long long tile = gwave;
    int buf = 0;
    if (tile < nTiles)
        tdm_issue_tile(baseP + (uint64_t)tile * TILE_BYTES,
                       baseT + (uint64_t)tile * TILE_BYTES,
                       ldsP[0], ldsT[0], g1);

    for (; tile < nTiles; tile += stride) {
        long long nxt = tile + stride;
        if (nxt < nTiles) {
            // buffer buf^1 was fully consumed last iteration; ensure our LDS
            // reads retired before the TDM engine overwrites it
            asm volatile("s_wait_dscnt 0" ::: "memory");
            tdm_issue_tile(baseP + (uint64_t)nxt * TILE_BYTES,
                           baseT + (uint64_t)nxt * TILE_BYTES,
                           ldsP[buf ^ 1], ldsT[buf ^ 1], g1);
            // 2 newly-issued TDM ops outstanding => both for 'tile' are done
            asm volatile("s_wait_tensorcnt 2" ::: "memory");
        } else {
            asm volatile("s_wait_tensorcnt 0" ::: "memory");
        }
        compute_tile(&tiles[wave][buf][0], &tiles[wave][buf][TILE_FLOATS],
                     lane, accM, accP, accA);
        buf ^= 1;
    }

    // wave32 reduction
#pragma unroll
    for (int off = 16; off > 0; off >>= 1) {
        accM += __shfl_down(accM, off, 32);
        accP += __shfl_down(accP, off, 32);
        accA += __shfl_down(accA, off, 32);
    }
    if (lane == 0) { red[wave][0] = accM; red[wave][1] = accP; red[wave][2] = accA; }
    __syncthreads();
    if (tid < 3) {
        float s = 0.0f;
#pragma unroll
        for (int w = 0; w < WPB; ++w) s += red[w][tid];
        ws[blockIdx.x * 3 + tid] = s;   // per-block partials (deterministic)
    }
}

// deterministic final reduction of NBLOCKS x 3 partials
__global__ __launch_bounds__(192) void pose_reduce(const float* __restrict__ ws,
                                                   float* __restrict__ out, int Bn) {
    __shared__ float s2[64 * 3];
    const int tid = threadIdx.x;          // 192 = 64 chunks * 3 metrics
    const int c = tid / 3, m = tid % 3;
    float s = 0.0f;
#pragma unroll
    for (int k = 0; k < NBLOCKS / 64; ++k) s += ws[(c * (NBLOCKS / 64) + k) * 3 + m];
    s2[c * 3 + m] = s;
    __syncthreads();
    if (tid < 3) {
        float tot = 0.0f;
        for (int cc = 0; cc < 64; ++cc) tot += s2[cc * 3 + tid];
        float denom = (tid == 2) ? (float)Bn * (float)(JOINTS - 2)
                                 : (float)Bn * (float)JOINTS;
        out[tid] = tot / denom;
    }
}

extern "C" void kernel_launch(void* const* d_in, const int* in_sizes, int n_in,
                              void* d_out, int out_size, void* d_ws, size_t ws_size,
                              hipStream_t stream) {
    const float* pred = (const float*)d_in[0];
    const float* targ = (const float*)d_in[1];
    float* ws = (float*)d_ws;
    float* out = (float*)d_out;

    const int Bn = in_sizes[0] / SAMPLE_FLOATS;   // 524288
    const int nTiles = Bn / TILE;                 // 16384

    pose_main<<<dim3(NBLOCKS), dim3(BLOCK), 0, stream>>>(pred, targ, ws, nTiles);
    pose_reduce<<<dim3(1), dim3(192), 0, stream>>>(ws, out, Bn);
}